// SpectralGCN_61469571940543
// MI455X (gfx1250) — compile-verified
//
#include <hip/hip_runtime.h>
#include <hip/hip_bf16.h>

typedef __attribute__((ext_vector_type(2))) float v2f;
typedef __attribute__((ext_vector_type(8))) float v8f;

#define NN 100000

// ---------------- utility kernels ----------------

__global__ __launch_bounds__(256) void zero_kernel(float* __restrict__ p, int n) {
  int i = blockIdx.x * 256 + threadIdx.x;
  if (i < n) p[i] = 0.0f;
}

__global__ __launch_bounds__(256) void degree_kernel(const int* __restrict__ src,
                                                     float* __restrict__ deg, int E) {
  int e = blockIdx.x * 256 + threadIdx.x;
  if (e < E) atomicAdd(&deg[src[e]], 1.0f);
}

__global__ __launch_bounds__(256) void dinv_kernel(float* __restrict__ deg, int n) {
  int i = blockIdx.x * 256 + threadIdx.x;
  if (i < n) {
    float d = deg[i];
    deg[i] = (d > 0.0f) ? rsqrtf(d) : 0.0f;   // in-place deg -> deg^-1/2
  }
}

__global__ __launch_bounds__(256) void norm_kernel(const int* __restrict__ src,
                                                   const int* __restrict__ dst,
                                                   const float* __restrict__ dinv,
                                                   float* __restrict__ nrm, int E) {
  int e = blockIdx.x * 256 + threadIdx.x;
  if (e < E) nrm[e] = -dinv[src[e]] * dinv[dst[e]];
}

// out[dst] += norm * x[src], 64-wide rows; 16 threads/edge, float4 per thread
__global__ __launch_bounds__(256) void prop64_kernel(const float* __restrict__ xin,
                                                     float* __restrict__ xout,
                                                     const int* __restrict__ src,
                                                     const int* __restrict__ dst,
                                                     const float* __restrict__ nrm, int E) {
  int tid = blockIdx.x * 256 + threadIdx.x;
  int e = tid >> 4;
  if (e >= E) return;
  int f = (tid & 15) << 2;
  float w = nrm[e];
  const float4 xv = *reinterpret_cast<const float4*>(xin + (size_t)src[e] * 64 + f);
  float* o = xout + (size_t)dst[e] * 64 + f;
  atomicAdd(o + 0, xv.x * w);
  atomicAdd(o + 1, xv.y * w);
  atomicAdd(o + 2, xv.z * w);
  atomicAdd(o + 3, xv.w * w);
}

// scalar propagate: out[dst] += norm * v[src]
__global__ __launch_bounds__(256) void prop1_kernel(const float* __restrict__ vin,
                                                    float* __restrict__ vout,
                                                    const int* __restrict__ src,
                                                    const int* __restrict__ dst,
                                                    const float* __restrict__ nrm, int E) {
  int e = blockIdx.x * 256 + threadIdx.x;
  if (e < E) atomicAdd(&vout[dst[e]], nrm[e] * vin[src[e]]);
}

// Wcat[192][64]: rows 0..63 = W1[0]-W1[2]; 64..127 = W1[1]; 128..191 = 2*W1[2]
__global__ __launch_bounds__(256) void wcat_kernel(const float* __restrict__ W1,
                                                   float* __restrict__ Wcat) {
  int i = blockIdx.x * 256 + threadIdx.x;
  if (i >= 192 * 64) return;
  int k = i >> 6, c = i & 63;
  float v;
  if (k < 64)       v = W1[k * 64 + c] - W1[2 * 4096 + k * 64 + c];
  else if (k < 128) v = W1[4096 + (k - 64) * 64 + c];
  else              v = 2.0f * W1[2 * 4096 + (k - 128) * 64 + c];
  Wcat[i] = v;
}

// ---------------- WMMA GEMM: h = relu([x|Tx1|S] @ Wcat + b1) ----------------
// Block = 256 threads = 8 waves; each wave produces a 16x64 output strip
// (one 16-row slab x all 4 N-tiles). K = 192 in steps of 4 via
// v_wmma_f32_16x16x4_f32 (fp32 A/B -> exact precision). K loop is split into
// 3 phases (x / Tx1 / S) so each phase has a loop-invariant A row pointer and
// A loads become immediate-offset global_load_b64.
__global__ __launch_bounds__(256) void gemm1_kernel(const float* __restrict__ x,
                                                    const float* __restrict__ Tx1,
                                                    const float* __restrict__ S,
                                                    const float* __restrict__ Wcat,
                                                    const float* __restrict__ bias,
                                                    float* __restrict__ h) {
  // K-paired LDS layout: sW[(k>>1)*128 + n*2 + (k&1)] so a B fragment
  // (K=kb,kb+1 for a given n) is one contiguous 8-byte LDS load.
  __shared__ float sW[192 * 64];
  for (int idx = threadIdx.x; idx < 192 * 64; idx += 256) {
    int k = idx >> 6, n = idx & 63;
    sW[(k >> 1) * 128 + (n << 1) + (k & 1)] = Wcat[idx];
  }
  __syncthreads();

  const int lane  = threadIdx.x & 31;
  const int wave  = threadIdx.x >> 5;
  const int m0    = blockIdx.x * 128 + wave * 16;
  const int nl    = lane & 15;
  const int khalf = (lane >> 4) << 1;            // 0 or 2
  const int mload = min(m0 + nl, NN - 1);        // clamp so EXEC stays all-1s

  v8f acc[4] = {v8f{}, v8f{}, v8f{}, v8f{}};

  const float* srcs[3] = {x, Tx1, S};
#pragma unroll
  for (int ph = 0; ph < 3; ++ph) {
    // loop-invariant A row pointer for this 64-wide K band
    const float* rowp = srcs[ph] + (size_t)mload * 64 + khalf;
    // loop-invariant LDS base for this band (pair-row of k = ph*64 + khalf)
    const float* wbase = &sW[((ph * 64 + khalf) >> 1) * 128];
#pragma unroll
    for (int kk = 0; kk < 16; ++kk) {
      // A fragment: (M=mload, K=kb) and (M=mload, K=kb+1), kb = ph*64+4*kk+khalf
      const float2 af = *reinterpret_cast<const float2*>(rowp + (kk << 2));
      v2f a; a.x = af.x; a.y = af.y;
      const float* wrow = wbase + kk * 256;      // advance 2 pair-rows per step
#pragma unroll
      for (int t = 0; t < 4; ++t) {
        const int n = (t << 4) + nl;
        const float2 bf = *reinterpret_cast<const float2*>(wrow + (n << 1));
        v2f b; b.x = bf.x; b.y = bf.y;
        acc[t] = __builtin_amdgcn_wmma_f32_16x16x4_f32(
            /*neg_a=*/false, a, /*neg_b=*/false, b,
            /*c_mod=*/(short)0, acc[t], /*reuse_a=*/false, /*reuse_b=*/false);
      }
    }
  }

  // epilogue: bias + relu + store (C/D layout: row = m0 + (lane/16)*8 + j)
  const int mbase = m0 + ((lane >> 4) << 3);
  if (m0 + 16 <= NN) {
    // full tile (781 of 782 blocks): guard-free coalesced-ish stores
    float* hp = h + (size_t)mbase * 64 + nl;
#pragma unroll
    for (int t = 0; t < 4; ++t) {
      const float bn = bias[(t << 4) + nl];
#pragma unroll
      for (int j = 0; j < 8; ++j)
        hp[(size_t)j * 64 + (t << 4)] = fmaxf(acc[t][j] + bn, 0.0f);
    }
  } else {
#pragma unroll
    for (int t = 0; t < 4; ++t) {
      const int n = (t << 4) + nl;
      const float bn = bias[n];
#pragma unroll
      for (int j = 0; j < 8; ++j) {
        const int m = mbase + j;
        if (m < NN) h[(size_t)m * 64 + n] = fmaxf(acc[t][j] + bn, 0.0f);
      }
    }
  }
}

// v_k[i] = sum_c h[i][c] * W2[k][c]  (layer-2 weights pushed through first)
__global__ __launch_bounds__(256) void vk_kernel(const float* __restrict__ h,
                                                 const float* __restrict__ W2,
                                                 float* __restrict__ v0,
                                                 float* __restrict__ v1,
                                                 float* __restrict__ v2) {
  int i = blockIdx.x * 256 + threadIdx.x;
  if (i >= NN) return;
  const float* hr = h + (size_t)i * 64;
  float a0 = 0.f, a1 = 0.f, a2 = 0.f;
#pragma unroll
  for (int c = 0; c < 64; ++c) {
    float hv = hr[c];
    a0 = fmaf(hv, W2[c], a0);
    a1 = fmaf(hv, W2[64 + c], a1);
    a2 = fmaf(hv, W2[128 + c], a2);
  }
  v0[i] = a0; v1[i] = a1; v2[i] = a2;
}

// out = v0 + P v1 + 2 P(P v2) - v2 + b2
__global__ __launch_bounds__(256) void final_kernel(const float* __restrict__ v0,
                                                    const float* __restrict__ v2,
                                                    const float* __restrict__ p1,
                                                    const float* __restrict__ r,
                                                    const float* __restrict__ b2,
                                                    float* __restrict__ out) {
  int i = blockIdx.x * 256 + threadIdx.x;
  if (i < NN) out[i] = v0[i] + p1[i] + 2.0f * r[i] - v2[i] + b2[0];
}

// ---------------- launcher ----------------

static inline int cdiv(long a, long b) { return (int)((a + b - 1) / b); }

extern "C" void kernel_launch(void* const* d_in, const int* in_sizes, int n_in,
                              void* d_out, int out_size, void* d_ws, size_t ws_size,
                              hipStream_t stream) {
  const float* x  = (const float*)d_in[0];
  const int*   ei = (const int*)d_in[1];
  const float* W1 = (const float*)d_in[2];
  const float* b1 = (const float*)d_in[3];
  const float* W2 = (const float*)d_in[4];
  const float* b2 = (const float*)d_in[5];
  const int E = in_sizes[1] / 2;
  const int* src = ei;
  const int* dst = ei + E;
  float* out = (float*)d_out;

  // workspace carve-up (floats): ~86 MB total
  float* w = (float*)d_ws;
  size_t off = 0;
  float* deg = w + off; off += NN;                 // deg -> dinv (in place)
  float* nrm = w + off; off += (size_t)E;          // per-edge norm
  float* Tx1 = w + off; off += (size_t)NN * 64;    // P x
  float* S   = w + off; off += (size_t)NN * 64;    // P Tx1
  float* h   = w + off; off += (size_t)NN * 64;    // layer-1 output
  float* Wc  = w + off; off += 192 * 64;           // combined layer-1 weights
  float* v0  = w + off; off += NN;
  float* v1  = w + off; off += NN;
  float* v2  = w + off; off += NN;
  float* p1  = w + off; off += NN;                 // P v1
  float* qq  = w + off; off += NN;                 // P v2
  float* rr  = w + off; off += NN;                 // P qq

  // zero all atomic accumulators every call (graph-replay safe)
  zero_kernel<<<cdiv(NN, 256), 256, 0, stream>>>(deg, NN);
  zero_kernel<<<cdiv((long)NN * 128, 256), 256, 0, stream>>>(Tx1, NN * 128); // Tx1+S contiguous
  zero_kernel<<<cdiv(3L * NN, 256), 256, 0, stream>>>(p1, 3 * NN);           // p1+qq+rr contiguous

  // normalization: deg -> dinv -> per-edge norm
  degree_kernel<<<cdiv(E, 256), 256, 0, stream>>>(src, deg, E);
  dinv_kernel<<<cdiv(NN, 256), 256, 0, stream>>>(deg, NN);
  norm_kernel<<<cdiv(E, 256), 256, 0, stream>>>(src, dst, deg, nrm, E);

  // Chebyshev propagations for layer 1 (bandwidth-bound, L2-resident)
  prop64_kernel<<<cdiv((long)E * 16, 256), 256, 0, stream>>>(x, Tx1, src, dst, nrm, E);
  prop64_kernel<<<cdiv((long)E * 16, 256), 256, 0, stream>>>(Tx1, S, src, dst, nrm, E);

  // fused layer-1 GEMM (WMMA fp32) + bias + relu
  wcat_kernel<<<cdiv(192 * 64, 256), 256, 0, stream>>>(W1, Wc);
  gemm1_kernel<<<cdiv(NN, 128), 256, 0, stream>>>(x, Tx1, S, Wc, b1, h);

  // layer 2: reduce to scalars first, then three cheap scalar propagates
  vk_kernel<<<cdiv(NN, 256), 256, 0, stream>>>(h, W2, v0, v1, v2);
  prop1_kernel<<<cdiv(E, 256), 256, 0, stream>>>(v1, p1, src, dst, nrm, E);
  prop1_kernel<<<cdiv(E, 256), 256, 0, stream>>>(v2, qq, src, dst, nrm, E);
  prop1_kernel<<<cdiv(E, 256), 256, 0, stream>>>(qq, rr, src, dst, nrm, E);

  final_kernel<<<cdiv(NN, 256), 256, 0, stream>>>(v0, v2, p1, rr, b2, out);
}